// wholeModel_58162447123369
// MI455X (gfx1250) — compile-verified
//
#include <hip/hip_runtime.h>
#include <hip/hip_bf16.h>

// ---------------- model dims ----------------
#define SEGN 32
#define FEAT 4096
#define ATTRN 300
#define NCN 51
#define BN_ 128
#define HN 256
#define NPAIR (BN_ * NCN)   // 6528
#define EPSV 1e-5f

typedef float v2f __attribute__((ext_vector_type(2)));
typedef float v8f __attribute__((ext_vector_type(8)));

// =====================================================================
// Generic f32 WMMA GEMM:  C[M,N] = act( A' @ B' + bias )
//   A' row r = (rowmap ? (rowmap[r]>=0 ? A[rowmap[r]] : 0) : A[r])  [lda]
//   B'(k,n)  = bTrans ? B[n*ldb+k] : B[k*ldb+n]
// Block tile 64x64, 256 threads (8 waves, 2 16x16 tiles each), K-chunk 16.
// =====================================================================
__global__ __launch_bounds__(256) void gemm_wmma_f32(
    const float* __restrict__ A, int lda, const int* __restrict__ rowmap,
    const float* __restrict__ B, int ldb, int bTrans,
    const float* __restrict__ bias, float* __restrict__ C, int ldc,
    int M, int N, int K, int act)
{
    __shared__ float As[64][17];   // padded: stride 17 avoids bank conflicts
    __shared__ float Bs[16][65];

    int tid  = threadIdx.x;
    int wave = tid >> 5, lane = tid & 31;
    int half = lane >> 4, lr = lane & 15;
    int m0 = blockIdx.y * 64, n0 = blockIdx.x * 64;

    int tm  = wave & 3;           // tile row 0..3
    int tn0 = (wave >> 2) << 1;   // tile cols {tn0, tn0+1}

    v8f c0 = {}; v8f c1 = {};

    for (int k0 = 0; k0 < K; k0 += 16) {
        for (int i = tid; i < 64 * 16; i += 256) {
            int r = i >> 4, kk = i & 15;
            int gr = m0 + r, gk = k0 + kk;
            float v = 0.f;
            if (gr < M && gk < K) {
                int src = rowmap ? rowmap[gr] : gr;
                if (src >= 0) v = A[(long)src * lda + gk];
            }
            As[r][kk] = v;
        }
        for (int i = tid; i < 16 * 64; i += 256) {
            int kk = i >> 6, cc = i & 63;
            int gk = k0 + kk, gc = n0 + cc;
            float v = 0.f;
            if (gk < K && gc < N)
                v = bTrans ? B[(long)gc * ldb + gk] : B[(long)gk * ldb + gc];
            Bs[kk][cc] = v;
        }
        __syncthreads();
        #pragma unroll
        for (int kk = 0; kk < 16; kk += 4) {
            v2f a;
            a.x = As[tm * 16 + lr][kk + 2 * half];
            a.y = As[tm * 16 + lr][kk + 2 * half + 1];
            v2f b0, b1;
            b0.x = Bs[kk + 2 * half][tn0 * 16 + lr];
            b0.y = Bs[kk + 2 * half + 1][tn0 * 16 + lr];
            b1.x = Bs[kk + 2 * half][tn0 * 16 + 16 + lr];
            b1.y = Bs[kk + 2 * half + 1][tn0 * 16 + 16 + lr];
            c0 = __builtin_amdgcn_wmma_f32_16x16x4_f32(false, a, false, b0, (short)0, c0, false, false);
            c1 = __builtin_amdgcn_wmma_f32_16x16x4_f32(false, a, false, b1, (short)0, c1, false, false);
        }
        __syncthreads();
    }

    int col0 = n0 + tn0 * 16 + lr;
    int col1 = col0 + 16;
    #pragma unroll
    for (int i = 0; i < 8; ++i) {
        int r = m0 + tm * 16 + i + 8 * half;
        if (r < M) {
            if (col0 < N) {
                float v = c0[i] + (bias ? bias[col0] : 0.f);
                if (act == 1) v = fmaxf(v, 0.f);
                C[(long)r * ldc + col0] = v;
            }
            if (col1 < N) {
                float v = c1[i] + (bias ? bias[col1] : 0.f);
                if (act == 1) v = fmaxf(v, 0.f);
                C[(long)r * ldc + col1] = v;
            }
        }
    }
}

// ---------------- attention layer 2 + sigmoid -> w[B*SEG] ----------------
__global__ void att2_kernel(const float* __restrict__ att1,
                            const float* __restrict__ w2,
                            const float* __restrict__ b2,
                            float* __restrict__ wout)
{
    int r = blockIdx.x * blockDim.x + threadIdx.x;
    if (r >= BN_ * SEGN) return;
    float s = b2[0];
    for (int k = 0; k < 256; ++k) s += att1[r * 256 + k] * w2[k];
    wout[r] = 1.f / (1.f + expf(-s));
}

// ---------------- stable segment compaction -> rowmap ----------------
__global__ void segsel_kernel(const float* __restrict__ w, int* __restrict__ rowmap)
{
    __shared__ float sw[SEGN];
    int b = blockIdx.x;
    if (threadIdx.x < SEGN) sw[threadIdx.x] = w[b * SEGN + threadIdx.x];
    __syncthreads();
    if (threadIdx.x == 0) {
        float m = 0.f;
        for (int i = 0; i < SEGN; ++i) m += sw[i];
        m /= (float)SEGN;
        int pos = 0;
        for (int i = 0; i < SEGN; ++i)
            if (sw[i] >= m) rowmap[b * SEGN + pos++] = b * SEGN + i;
        for (; pos < SEGN; ++pos) rowmap[b * SEGN + pos] = -1;
    }
}

// ---------------- per-column BatchNorm (train stats), in place ----------------
__global__ void bn_cols_kernel(float* __restrict__ x, const float* __restrict__ g,
                               const float* __restrict__ bta, int R, int C)
{
    int c = blockIdx.x * blockDim.x + threadIdx.x;
    if (c >= C) return;
    float s = 0.f, s2 = 0.f;
    for (int r = 0; r < R; ++r) { float v = x[(long)r * C + c]; s += v; s2 += v * v; }
    float mean = s / (float)R;
    float var = s2 / (float)R - mean * mean;
    float inv = rsqrtf(var + EPSV);
    for (int r = 0; r < R; ++r)
        x[(long)r * C + c] = (x[(long)r * C + c] - mean) * inv * g[c] + bta[c];
}

// =====================================================================
// Bidirectional GRU recurrence. blockIdx.x = direction (0 fwd, 1 bwd).
// h [32,256] lives in LDS; per step hg = h @ whh^T + bhh via WMMA,
// fused gate update, writes gru_out[t,seg, dir*256 + j].
// =====================================================================
__global__ __launch_bounds__(1024) void gru_kernel(
    const float* __restrict__ xg_f, const float* __restrict__ xg_b,
    const float* __restrict__ whh_f, const float* __restrict__ bhh_f,
    const float* __restrict__ whh_b, const float* __restrict__ bhh_b,
    float* __restrict__ out)
{
    extern __shared__ float sh[];
    const int HS = 257, HGS = 769;
    float* hbuf = sh;               // [32][257]
    float* hg   = sh + 32 * HS;     // [32][769]

    int dir = blockIdx.x;
    const float* xg  = dir ? xg_b  : xg_f;
    const float* whh = dir ? whh_b : whh_f;
    const float* bhh = dir ? bhh_b : bhh_f;

    int tid = threadIdx.x, wave = tid >> 5, lane = tid & 31;
    int half = lane >> 4, lr = lane & 15;

    for (int i = tid; i < 32 * HN; i += 1024)
        hbuf[(i >> 8) * HS + (i & 255)] = 0.f;
    __syncthreads();

    for (int step = 0; step < 128; ++step) {
        int t = dir ? 127 - step : step;
        // hg = h @ whh^T + bhh : tiles 2 x 48 = 96, 32 waves -> 3 each
        for (int tile = wave; tile < 96; tile += 32) {
            int tmr = tile & 1, tnc = tile >> 1;
            int col = tnc * 16 + lr;
            const float* bcol = whh + (long)col * HN;   // whh[col][k]
            v8f c = {};
            #pragma unroll 4
            for (int k = 0; k < HN; k += 4) {
                v2f a, bb;
                a.x  = hbuf[(tmr * 16 + lr) * HS + k + 2 * half];
                a.y  = hbuf[(tmr * 16 + lr) * HS + k + 2 * half + 1];
                bb.x = bcol[k + 2 * half];
                bb.y = bcol[k + 2 * half + 1];
                c = __builtin_amdgcn_wmma_f32_16x16x4_f32(false, a, false, bb, (short)0, c, false, false);
            }
            float bv = bhh[col];
            #pragma unroll
            for (int i = 0; i < 8; ++i)
                hg[(tmr * 16 + i + 8 * half) * HGS + col] = c[i] + bv;
        }
        __syncthreads();
        // fused gate update (xg already holds x@wih^T + bih)
        const float* xgt = xg + (long)t * 32 * 768;
        for (int i = tid; i < 32 * HN; i += 1024) {
            int n = i >> 8, j = i & 255;
            float hr = hg[n * HGS + j];
            float hz = hg[n * HGS + HN + j];
            float hn = hg[n * HGS + 2 * HN + j];
            float xr = xgt[n * 768 + j];
            float xz = xgt[n * 768 + HN + j];
            float xn = xgt[n * 768 + 2 * HN + j];
            float r  = 1.f / (1.f + expf(-(xr + hr)));
            float z  = 1.f / (1.f + expf(-(xz + hz)));
            float nn = tanhf(xn + r * hn);
            float hp = hbuf[n * HS + j];
            float hv = (1.f - z) * nn + z * hp;
            hbuf[n * HS + j] = hv;
            out[((long)t * 32 + n) * 512 + dir * HN + j] = hv;
        }
        __syncthreads();
    }
}

// ---------------- zero small buffer ----------------
__global__ void zero_kernel(float* __restrict__ p, int n)
{
    int i = blockIdx.x * blockDim.x + threadIdx.x;
    if (i < n) p[i] = 0.f;
}

// =====================================================================
// Relation branch: one block per (b,c) pair.
//   sq = (gru_out[b] - sa[c])^2 in LDS, x1 = relu(sq @ fc3 + b3) via WMMA.
//   mode 0: accumulate per-seg sum/sumsq (BN4 stats, atomics)
//   mode 1: BN4-normalize + fc4 dot -> y[pair, seg]
// =====================================================================
__global__ __launch_bounds__(256) void relation_kernel(
    const float* __restrict__ gru, const float* __restrict__ sa,
    const float* __restrict__ fc3w, const float* __restrict__ fc3b,
    const float* __restrict__ fc4w, const float* __restrict__ fc4b,
    const float* __restrict__ bn4g, const float* __restrict__ bn4b,
    float* __restrict__ bn4sum, float* __restrict__ bn4sq,
    float* __restrict__ y, int mode)
{
    extern __shared__ float sh[];
    const int SQS = 513, XS = 257, BS = 257;
    float* sav = sh;                  // 512
    float* sq  = sav + 512;           // 32*513
    float* Bs  = sq + 32 * SQS;       // 16*257
    float* x1  = Bs + 16 * BS;        // 32*257
    float* red = x1 + 32 * XS;        // 512

    int nIdx = blockIdx.x;
    int b = nIdx / NCN, c = nIdx % NCN;
    int tid = threadIdx.x, wave = tid >> 5, lane = tid & 31;
    int half = lane >> 4, lr = lane & 15;

    for (int i = tid; i < 512; i += 256) sav[i] = sa[c * 512 + i];
    __syncthreads();
    const float* gb = gru + (long)b * 32 * 512;
    for (int i = tid; i < 32 * 512; i += 256) {
        int n = i >> 9, k = i & 511;
        float d = gb[n * 512 + k] - sav[k];
        sq[n * SQS + k] = d * d;
    }

    int tmr = wave & 1, tnc = wave >> 1;     // 4 tiles/wave: cols (tnc+4j)*16
    v8f acc[4] = {};
    for (int k0 = 0; k0 < 512; k0 += 16) {
        __syncthreads();
        for (int i = tid; i < 16 * 256; i += 256) {
            int kk = i >> 8, cc = i & 255;
            Bs[kk * BS + cc] = fc3w[(k0 + kk) * 256 + cc];
        }
        __syncthreads();
        #pragma unroll
        for (int kk = 0; kk < 16; kk += 4) {
            v2f a;
            a.x = sq[(tmr * 16 + lr) * SQS + k0 + kk + 2 * half];
            a.y = sq[(tmr * 16 + lr) * SQS + k0 + kk + 2 * half + 1];
            #pragma unroll
            for (int j = 0; j < 4; ++j) {
                int col = (tnc + 4 * j) * 16 + lr;
                v2f bb;
                bb.x = Bs[(kk + 2 * half) * BS + col];
                bb.y = Bs[(kk + 2 * half + 1) * BS + col];
                acc[j] = __builtin_amdgcn_wmma_f32_16x16x4_f32(false, a, false, bb, (short)0, acc[j], false, false);
            }
        }
    }
    __syncthreads();
    #pragma unroll
    for (int j = 0; j < 4; ++j) {
        int col = (tnc + 4 * j) * 16 + lr;
        float bv = fc3b[col];
        #pragma unroll
        for (int i = 0; i < 8; ++i)
            x1[(tmr * 16 + i + 8 * half) * XS + col] = fmaxf(acc[j][i] + bv, 0.f);
    }
    __syncthreads();

    if (mode == 0) {
        int row = tid >> 3, p = tid & 7;     // 32 rows x 8 partials
        float s = 0.f, s2 = 0.f;
        for (int cc = p * 32; cc < p * 32 + 32; ++cc) {
            float v = x1[row * XS + cc]; s += v; s2 += v * v;
        }
        red[tid] = s; red[256 + tid] = s2;
        __syncthreads();
        if (tid < 32) {
            float ts = 0.f, ts2 = 0.f;
            for (int p2 = 0; p2 < 8; ++p2) { ts += red[tid * 8 + p2]; ts2 += red[256 + tid * 8 + p2]; }
            atomicAdd(&bn4sum[tid], ts);
            atomicAdd(&bn4sq[tid], ts2);
        }
    } else {
        if (tid < 32) {
            float cnt = (float)NPAIR * 256.f;
            float mean = bn4sum[tid] / cnt;
            float var = bn4sq[tid] / cnt - mean * mean;
            red[tid] = mean;
            red[32 + tid] = rsqrtf(var + EPSV);
        }
        __syncthreads();
        for (int row = wave; row < 32; row += 8) {
            float mean = red[row], inv = red[32 + row];
            float g = bn4g[row], bb = bn4b[row];
            float s = 0.f;
            for (int cc = lane; cc < 256; cc += 32) {
                float v = (x1[row * XS + cc] - mean) * inv * g + bb;
                s += v * fc4w[cc];
            }
            for (int off = 16; off > 0; off >>= 1) s += __shfl_down(s, off, 32);
            if (lane == 0) y[nIdx * 32 + row] = s + fc4b[0];
        }
    }
}

// ---------------- per-segment softmax stats ----------------
__global__ void segstat_kernel(const float* __restrict__ y,
                               float* __restrict__ segmax, float* __restrict__ segsum)
{
    __shared__ float red[256];
    int s = blockIdx.x, tid = threadIdx.x;
    float m = -1e30f;
    for (int n = tid; n < NPAIR; n += 256) m = fmaxf(m, y[n * 32 + s]);
    red[tid] = m; __syncthreads();
    for (int off = 128; off > 0; off >>= 1) {
        if (tid < off) red[tid] = fmaxf(red[tid], red[tid + off]);
        __syncthreads();
    }
    float mx = red[0]; __syncthreads();
    float sum = 0.f;
    for (int n = tid; n < NPAIR; n += 256) sum += expf(y[n * 32 + s] - mx);
    red[tid] = sum; __syncthreads();
    for (int off = 128; off > 0; off >>= 1) {
        if (tid < off) red[tid] += red[tid + off];
        __syncthreads();
    }
    if (tid == 0) { segmax[s] = mx; segsum[s] = red[0]; }
}

// ---------------- final: softmax over pairs, mean over segs ----------------
__global__ void final_kernel(const float* __restrict__ y,
                             const float* __restrict__ segmax,
                             const float* __restrict__ segsum,
                             float* __restrict__ out)
{
    int n = blockIdx.x * blockDim.x + threadIdx.x;
    if (n >= NPAIR) return;
    float acc = 0.f;
    for (int s = 0; s < 32; ++s)
        acc += expf(y[n * 32 + s] - segmax[s]) / segsum[s];
    out[n] = acc / 32.f;
}

// =====================================================================
extern "C" void kernel_launch(void* const* d_in, const int* in_sizes, int n_in,
                              void* d_out, int out_size, void* d_ws, size_t ws_size,
                              hipStream_t stream)
{
    const float* bf    = (const float*)d_in[0];   // [128,32,4096]
    const float* attrs = (const float*)d_in[1];   // [51,300]
    const float* a_w1  = (const float*)d_in[2];
    const float* a_b1  = (const float*)d_in[3];
    const float* a_w2  = (const float*)d_in[4];
    const float* a_b2  = (const float*)d_in[5];
    const float* fc1_w = (const float*)d_in[6];
    const float* fc1_b = (const float*)d_in[7];
    const float* bn1_g = (const float*)d_in[8];
    const float* bn1_b = (const float*)d_in[9];
    const float* fc2_w = (const float*)d_in[10];
    const float* fc2_b = (const float*)d_in[11];
    const float* bn2_g = (const float*)d_in[12];
    const float* bn2_b = (const float*)d_in[13];
    const float* wih_f = (const float*)d_in[14];
    const float* whh_f = (const float*)d_in[15];
    const float* bih_f = (const float*)d_in[16];
    const float* bhh_f = (const float*)d_in[17];
    const float* wih_b = (const float*)d_in[18];
    const float* whh_b = (const float*)d_in[19];
    const float* bih_b = (const float*)d_in[20];
    const float* bhh_b = (const float*)d_in[21];
    const float* fc3_w = (const float*)d_in[22];
    const float* fc3_b = (const float*)d_in[23];
    const float* bn4_g = (const float*)d_in[24];
    const float* bn4_b = (const float*)d_in[25];
    const float* fc4_w = (const float*)d_in[26];
    const float* fc4_b = (const float*)d_in[27];
    float* out = (float*)d_out;

    float* ws = (float*)d_ws;
    size_t o = 0;
    float* w_att  = ws + o; o += BN_ * SEGN;                 // 4096
    int*   rowmap = (int*)(ws + o); o += BN_ * SEGN;         // 4096
    float* att1   = ws + o; o += (size_t)BN_ * SEGN * 256;   // 1M
    float* xgf    = ws + o; o += (size_t)BN_ * SEGN * 768;   // 3.1M
    float* xgb    = ws + o; o += (size_t)BN_ * SEGN * 768;   // 3.1M
    float* gruo   = ws + o; o += (size_t)BN_ * SEGN * 512;   // 2.1M
    float* sa1    = ws + o; o += (size_t)NCN * FEAT;         // 209K
    float* sa2    = ws + o; o += (size_t)NCN * 512;          // 26K
    float* bn4sum = ws + o; o += 32;
    float* bn4sq  = ws + o; o += 32;
    float* ybuf   = ws + o; o += (size_t)NPAIR * 32;         // 209K
    float* segmax = ws + o; o += 32;
    float* segsum = ws + o; o += 32;

    const int M_BS = BN_ * SEGN;   // 4096

    // 1) attention layer 1: att1 = relu(bf @ a_w1 + a_b1)   [4096,256]
    gemm_wmma_f32<<<dim3(256 / 64, M_BS / 64), 256, 0, stream>>>(
        bf, FEAT, nullptr, a_w1, 256, 0, a_b1, att1, 256, M_BS, 256, FEAT, 1);
    // 2) attention layer 2 + sigmoid -> w
    att2_kernel<<<(M_BS + 255) / 256, 256, 0, stream>>>(att1, a_w2, a_b2, w_att);
    // 3) stable compaction -> rowmap (-1 = zero row)
    segsel_kernel<<<BN_, 32, 0, stream>>>(w_att, rowmap);
    // 4) GRU input projections: xg = valid @ wih^T + bih   [4096,768]
    gemm_wmma_f32<<<dim3(768 / 64, M_BS / 64), 256, 0, stream>>>(
        bf, FEAT, rowmap, wih_f, FEAT, 1, bih_f, xgf, 768, M_BS, 768, FEAT, 0);
    gemm_wmma_f32<<<dim3(768 / 64, M_BS / 64), 256, 0, stream>>>(
        bf, FEAT, rowmap, wih_b, FEAT, 1, bih_b, xgb, 768, M_BS, 768, FEAT, 0);
    // 5) GRU recurrence (fwd+bwd), fused -> gruo [128,32,512]
    {
        size_t shb = (size_t)(32 * 257 + 32 * 769) * sizeof(float);
        gru_kernel<<<2, 1024, shb, stream>>>(xgf, xgb, whh_f, bhh_f, whh_b, bhh_b, gruo);
    }
    // 6) semantic branch
    gemm_wmma_f32<<<dim3(FEAT / 64, 1), 256, 0, stream>>>(
        attrs, ATTRN, nullptr, fc1_w, FEAT, 0, fc1_b, sa1, FEAT, NCN, FEAT, ATTRN, 1);
    bn_cols_kernel<<<(FEAT + 255) / 256, 256, 0, stream>>>(sa1, bn1_g, bn1_b, NCN, FEAT);
    gemm_wmma_f32<<<dim3(512 / 64, 1), 256, 0, stream>>>(
        sa1, FEAT, nullptr, fc2_w, 512, 0, fc2_b, sa2, 512, NCN, 512, FEAT, 1);
    bn_cols_kernel<<<(512 + 255) / 256, 256, 0, stream>>>(sa2, bn2_g, bn2_b, NCN, 512);
    // 7) relation branch: stats pass, then normalize+fc4 pass
    zero_kernel<<<1, 64, 0, stream>>>(bn4sum, 64);   // zeroes bn4sum+bn4sq (contiguous)
    {
        size_t shb = (size_t)(512 + 32 * 513 + 16 * 257 + 32 * 257 + 512) * sizeof(float);
        relation_kernel<<<NPAIR, 256, shb, stream>>>(
            gruo, sa2, fc3_w, fc3_b, fc4_w, fc4_b, bn4_g, bn4_b,
            bn4sum, bn4sq, ybuf, 0);
        relation_kernel<<<NPAIR, 256, shb, stream>>>(
            gruo, sa2, fc3_w, fc3_b, fc4_w, fc4_b, bn4_g, bn4_b,
            bn4sum, bn4sq, ybuf, 1);
    }
    // 8) softmax over pairs (axis 0), mean over segments
    segstat_kernel<<<32, 256, 0, stream>>>(ybuf, segmax, segsum);
    final_kernel<<<(NPAIR + 255) / 256, 256, 0, stream>>>(ybuf, segmax, segsum, out);
}